// GlobalExpertPool_31662498906566
// MI455X (gfx1250) — compile-verified
//
#include <hip/hip_runtime.h>
#include <hip/hip_bf16.h>

typedef __attribute__((ext_vector_type(2))) float v2f;
typedef __attribute__((ext_vector_type(4))) float v4f;
typedef __attribute__((ext_vector_type(8))) float v8f;

#define T_TOK   8192
#define DIN     4096
#define DOUT    4096
#define NEXP    8
#define NR      8
#define NC      64          // NEXP * NR
#define KSPLIT  8
#define SCALING 2.0f        // 16 / r

// ---------------- workspace layout (floats) ----------------
// [0..8)            probsum_a
// [8..16)           probsum_b
// [16..16+T*8)      coef_a
// [..+T*8)          coef_b
// [..+T*64)         a_all  (atomically accumulated; rewritten in-place as mid_e)
#define WS_PSA   0
#define WS_PSB   8
#define WS_CA    16
#define WS_CB    (WS_CA + T_TOK * NEXP)
#define WS_AALL  (WS_CB + T_TOK * NEXP)
#define WS_FLOATS (WS_AALL + T_TOK * NC)

__global__ __launch_bounds__(256) void k_zero(float* __restrict__ p, int n) {
  int i = blockIdx.x * 256 + threadIdx.x;
  if (i < n) p[i] = 0.0f;
}

// ---------------- router: softmax sums (aux) + top-2 coefficients ----------------
__global__ __launch_bounds__(256) void k_router(const float* __restrict__ la,
                                                const float* __restrict__ lb,
                                                float* __restrict__ coef_a,
                                                float* __restrict__ coef_b,
                                                float* __restrict__ ps_a,
                                                float* __restrict__ ps_b) {
  __shared__ float sA[NEXP];
  __shared__ float sB[NEXP];
  if (threadIdx.x < NEXP) sA[threadIdx.x] = 0.0f;
  else if (threadIdx.x < 2 * NEXP) sB[threadIdx.x - NEXP] = 0.0f;
  __syncthreads();

  int t = blockIdx.x * 256 + threadIdx.x;
  if (t < T_TOK) {
    #pragma unroll
    for (int side = 0; side < 2; ++side) {
      const float* lg = (side == 0 ? la : lb) + (size_t)t * NEXP;
      float l[NEXP];
      float mx = -1e30f;
      #pragma unroll
      for (int e = 0; e < NEXP; ++e) { l[e] = lg[e]; mx = fmaxf(mx, l[e]); }
      float p[NEXP]; float sum = 0.0f;
      #pragma unroll
      for (int e = 0; e < NEXP; ++e) { p[e] = __expf(l[e] - mx); sum += p[e]; }
      float inv = 1.0f / sum;
      float* sh = side == 0 ? sA : sB;
      #pragma unroll
      for (int e = 0; e < NEXP; ++e) atomicAdd(&sh[e], p[e] * inv);
      // top-2 (first index wins ties, like lax.top_k)
      int i1 = 0; float b1 = l[0];
      #pragma unroll
      for (int e = 1; e < NEXP; ++e) if (l[e] > b1) { b1 = l[e]; i1 = e; }
      int i2 = (i1 == 0) ? 1 : 0; float b2 = l[i2];
      #pragma unroll
      for (int e = 0; e < NEXP; ++e)
        if (e != i1 && l[e] > b2) { b2 = l[e]; i2 = e; }
      float e2 = __expf(b2 - b1);
      float w1 = 1.0f / (1.0f + e2);
      float w2 = e2 * w1;
      float scale = side == 0 ? 1.0f : SCALING;
      float* cf = (side == 0 ? coef_a : coef_b) + (size_t)t * NEXP;
      #pragma unroll
      for (int e = 0; e < NEXP; ++e) cf[e] = 0.0f;
      cf[i1] = w1 * scale;
      cf[i2] = w2 * scale;
    }
  }
  __syncthreads();
  if (threadIdx.x < NEXP) atomicAdd(&ps_a[threadIdx.x], sA[threadIdx.x]);
  else if (threadIdx.x < 2 * NEXP) atomicAdd(&ps_b[threadIdx.x - NEXP], sB[threadIdx.x - NEXP]);
}

// ---------------- stage A GEMM: a_all[T,64] += x[T,4096] * WaT[4096,64] ----------------
// Wa layout [E,R,D] => WaT[d, n] = Wa[n*DIN + d] with n = e*8+r.
// One wave per 16-token tile per K-slice; V_WMMA_F32_16X16X4_F32.
__global__ __launch_bounds__(256) void k_gemm_a(const float* __restrict__ x,
                                                const float* __restrict__ Wa,
                                                float* __restrict__ a_all) {
  const int wave = threadIdx.x >> 5;
  const int lane = threadIdx.x & 31;
  const int lo = lane & 15, hi = lane >> 4;
  const int tile = blockIdx.x * 8 + wave;           // 0..511
  const int t0 = tile * 16;
  const int kbeg = blockIdx.y * (DIN / KSPLIT);
  const int kend = kbeg + (DIN / KSPLIT);

  v8f a0 = {}, a1 = {}, a2 = {}, a3 = {};
  const float* xrow = x  + (size_t)(t0 + lo) * DIN + 2 * hi;
  const float* w0   = Wa + (size_t)(lo +  0) * DIN + 2 * hi;
  const float* w1   = Wa + (size_t)(lo + 16) * DIN + 2 * hi;
  const float* w2   = Wa + (size_t)(lo + 32) * DIN + 2 * hi;
  const float* w3   = Wa + (size_t)(lo + 48) * DIN + 2 * hi;

  #pragma unroll 4
  for (int k0 = kbeg; k0 < kend; k0 += 4) {
    v2f av = *(const v2f*)(xrow + k0);
    v2f b0 = *(const v2f*)(w0 + k0);
    v2f b1 = *(const v2f*)(w1 + k0);
    v2f b2 = *(const v2f*)(w2 + k0);
    v2f b3 = *(const v2f*)(w3 + k0);
    a0 = __builtin_amdgcn_wmma_f32_16x16x4_f32(false, av, false, b0, (short)0, a0, false, false);
    a1 = __builtin_amdgcn_wmma_f32_16x16x4_f32(false, av, false, b1, (short)0, a1, false, false);
    a2 = __builtin_amdgcn_wmma_f32_16x16x4_f32(false, av, false, b2, (short)0, a2, false, false);
    a3 = __builtin_amdgcn_wmma_f32_16x16x4_f32(false, av, false, b3, (short)0, a3, false, false);
  }
  // C layout: vgpr v, lanes 0-15 -> m=v ; lanes 16-31 -> m=v+8 ; n = lane&15 (+16*j)
  #pragma unroll
  for (int v = 0; v < 8; ++v) {
    float* dst = a_all + (size_t)(t0 + v + 8 * hi) * NC + lo;
    atomicAdd(dst +  0, a0[v]);
    atomicAdd(dst + 16, a1[v]);
    atomicAdd(dst + 32, a2[v]);
    atomicAdd(dst + 48, a3[v]);
  }
}

// ---------------- per-token combine: a_all row -> mid_e row (in place) ----------------
__global__ __launch_bounds__(256) void k_combine(const float* __restrict__ coef_a,
                                                 const float* __restrict__ coef_b,
                                                 float* __restrict__ am) {
  int t = blockIdx.x * 256 + threadIdx.x;
  if (t >= T_TOK) return;
  float ca[NEXP], cb[NEXP], row[NC];
  #pragma unroll
  for (int e = 0; e < NEXP; ++e) { ca[e] = coef_a[(size_t)t * NEXP + e]; cb[e] = coef_b[(size_t)t * NEXP + e]; }
  float* rp = am + (size_t)t * NC;
  #pragma unroll
  for (int q = 0; q < NC / 4; ++q) { v4f v = *(const v4f*)(rp + 4 * q);
    row[4*q] = v.x; row[4*q+1] = v.y; row[4*q+2] = v.z; row[4*q+3] = v.w; }
  float mid[NR];
  #pragma unroll
  for (int r = 0; r < NR; ++r) {
    float s = 0.0f;
    #pragma unroll
    for (int e = 0; e < NEXP; ++e) s = fmaf(ca[e], row[e * NR + r], s);
    mid[r] = s;
  }
  #pragma unroll
  for (int e = 0; e < NEXP; ++e)
    #pragma unroll
    for (int r = 0; r < NR; ++r) row[e * NR + r] = cb[e] * mid[r];
  #pragma unroll
  for (int q = 0; q < NC / 4; ++q) {
    v4f v = { row[4*q], row[4*q+1], row[4*q+2], row[4*q+3] };
    *(v4f*)(rp + 4 * q) = v;
  }
}

// ---------------- stage B GEMM: out[T,4096] = mid_e[T,64] * WbT[64,4096] ----------------
// Wb layout [E,DOUT,R] => WbT[k, o] = Wb[(k>>3)*DOUT*8 + o*8 + (k&7)].
__global__ __launch_bounds__(256) void k_gemm_b(const float* __restrict__ mid_e,
                                                const float* __restrict__ Wb,
                                                float* __restrict__ out) {
  const int wave = threadIdx.x >> 5;
  const int lane = threadIdx.x & 31;
  const int lo = lane & 15, hi = lane >> 4;
  const int w = blockIdx.x * 8 + wave;              // 0..32767
  const int nt = w & 63, mt = w >> 6;
  const int t0 = mt * 16, o0 = nt * 64;

  // whole 16x64 A tile in registers
  v2f av[16];
  const float* arow = mid_e + (size_t)(t0 + lo) * NC + 2 * hi;
  #pragma unroll
  for (int kk = 0; kk < 16; ++kk) av[kk] = *(const v2f*)(arow + kk * 4);

  v8f a0 = {}, a1 = {}, a2 = {}, a3 = {};
  #pragma unroll
  for (int kk = 0; kk < 16; ++kk) {
    const int k = kk * 4 + 2 * hi;                  // even -> 8B aligned pairs
    const float* wb = Wb + (size_t)(k >> 3) * (DOUT * NR) + (k & 7);
    v2f b0 = *(const v2f*)(wb + (size_t)(o0 +  0 + lo) * NR);
    v2f b1 = *(const v2f*)(wb + (size_t)(o0 + 16 + lo) * NR);
    v2f b2 = *(const v2f*)(wb + (size_t)(o0 + 32 + lo) * NR);
    v2f b3 = *(const v2f*)(wb + (size_t)(o0 + 48 + lo) * NR);
    a0 = __builtin_amdgcn_wmma_f32_16x16x4_f32(false, av[kk], false, b0, (short)0, a0, false, false);
    a1 = __builtin_amdgcn_wmma_f32_16x16x4_f32(false, av[kk], false, b1, (short)0, a1, false, false);
    a2 = __builtin_amdgcn_wmma_f32_16x16x4_f32(false, av[kk], false, b2, (short)0, a2, false, false);
    a3 = __builtin_amdgcn_wmma_f32_16x16x4_f32(false, av[kk], false, b3, (short)0, a3, false, false);
  }
  #pragma unroll
  for (int v = 0; v < 8; ++v) {
    float* dst = out + (size_t)(t0 + v + 8 * hi) * DOUT + o0 + lo;
    dst[ 0] = a0[v];
    dst[16] = a1[v];
    dst[32] = a2[v];
    dst[48] = a3[v];
  }
}

// ---------------- aux losses ----------------
__global__ void k_aux(const float* __restrict__ ps_a, const float* __restrict__ ps_b,
                      float* __restrict__ out) {
  if (threadIdx.x == 0 && blockIdx.x == 0) {
    #pragma unroll
    for (int side = 0; side < 2; ++side) {
      const float* ps = side == 0 ? ps_a : ps_b;
      float p[NEXP]; float mean = 0.0f;
      #pragma unroll
      for (int e = 0; e < NEXP; ++e) { p[e] = ps[e] * (1.0f / T_TOK); mean += p[e]; }
      mean *= (1.0f / NEXP);
      float var = 0.0f;
      #pragma unroll
      for (int e = 0; e < NEXP; ++e) { float d = p[e] - mean; var += d * d; }
      var *= (1.0f / (NEXP - 1));                 // unbiased (ddof=1)
      out[(size_t)T_TOK * DOUT + side] = NEXP * var;
    }
  }
}

extern "C" void kernel_launch(void* const* d_in, const int* in_sizes, int n_in,
                              void* d_out, int out_size, void* d_ws, size_t ws_size,
                              hipStream_t stream) {
  const float* x  = (const float*)d_in[0];
  const float* la = (const float*)d_in[1];
  const float* lb = (const float*)d_in[2];
  const float* Wa = (const float*)d_in[3];
  const float* Wb = (const float*)d_in[4];
  float* out = (float*)d_out;
  float* ws  = (float*)d_ws;

  float* ps_a   = ws + WS_PSA;
  float* ps_b   = ws + WS_PSB;
  float* coef_a = ws + WS_CA;
  float* coef_b = ws + WS_CB;
  float* a_all  = ws + WS_AALL;   // becomes mid_e after k_combine

  // zero atomic accumulators (probsums + a_all)
  {
    int nz = 2 * NEXP + T_TOK * NC;
    // zero probsums (start of ws) and a_all separately to skip coef region
    k_zero<<<(2 * NEXP + 255) / 256, 256, 0, stream>>>(ws, 2 * NEXP);
    k_zero<<<(T_TOK * NC + 255) / 256, 256, 0, stream>>>(a_all, T_TOK * NC);
    (void)nz;
  }
  k_router<<<T_TOK / 256, 256, 0, stream>>>(la, lb, coef_a, coef_b, ps_a, ps_b);
  k_gemm_a<<<dim3((T_TOK / 16) / 8, KSPLIT), 256, 0, stream>>>(x, Wa, a_all);
  k_combine<<<T_TOK / 256, 256, 0, stream>>>(coef_a, coef_b, a_all);
  k_gemm_b<<<((T_TOK / 16) * (DOUT / 64)) / 8, 256, 0, stream>>>(a_all, Wb, out);
  k_aux<<<1, 32, 0, stream>>>(ps_a, ps_b, out);
  (void)in_sizes; (void)n_in; (void)out_size; (void)ws_size;
}